// ModelOT_68281390071988
// MI455X (gfx1250) — compile-verified
//
#include <hip/hip_runtime.h>
#include <hip/hip_bf16.h>

// ---------------------------------------------------------------------------
// ModelOT (Sinkhorn OT) for MI455X / gfx1250, wave32.
//   Phase 0: mean(D_tensor)                 (2 tree-reduction kernels)
//   Phase 1: K = exp(5*S[b,pois] - 5*Dn)    (WMMA f32 16x16x4 GEMM + LDS gather)
//            S-tile [16 x 4096] built per workgroup with V_WMMA_F32_16X16X4_F32
//            into 256KB LDS (CDNA5 WGP has 320KB).  K stored in d_out.
//   Phase 2: 10 Sinkhorn iterations, K L2-resident (64MB << 192MB L2).
//   Phase 3: P = K * u * v  in place in d_out.
// ---------------------------------------------------------------------------

#define BDIM 4096
#define NDIM 4096
#define DDIM 16

typedef __attribute__((ext_vector_type(2))) float v2f;
typedef __attribute__((ext_vector_type(8))) float v8f;

// ---------------- Phase 0: mean of D_tensor -------------------------------
__global__ void __launch_bounds__(256)
mean_partial_kernel(const float* __restrict__ Dt, float* __restrict__ partials)
{
    __shared__ float red[256];
    const long total  = (long)BDIM * NDIM;
    const long stride = (long)gridDim.x * 256;
    float sum = 0.0f;
    for (long i = (long)blockIdx.x * 256 + threadIdx.x; i < total; i += stride)
        sum += Dt[i];
    red[threadIdx.x] = sum;
    __syncthreads();
    for (int s = 128; s > 0; s >>= 1) {
        if (threadIdx.x < (unsigned)s) red[threadIdx.x] += red[threadIdx.x + s];
        __syncthreads();
    }
    if (threadIdx.x == 0) partials[blockIdx.x] = red[0];
}

__global__ void __launch_bounds__(256)
mean_final_kernel(const float* __restrict__ partials, float* __restrict__ mean_out)
{
    __shared__ float red[256];
    red[threadIdx.x] = partials[threadIdx.x];
    __syncthreads();
    for (int s = 128; s > 0; s >>= 1) {
        if (threadIdx.x < (unsigned)s) red[threadIdx.x] += red[threadIdx.x + s];
        __syncthreads();
    }
    if (threadIdx.x == 0)
        mean_out[0] = red[0] / (float)((long)BDIM * NDIM);
}

// ---------------- Phase 1: build K via WMMA + LDS gather ------------------
// One workgroup (256 thr = 8 waves) per 16-row b-tile.
// S_tile[16][4096] = A[16x16] (gathered user embeddings) @ poi_emb^T[16x4096]
// computed as 256 column tiles of 16x16, each = 4 chained V_WMMA_F32_16X16X4_F32.
__global__ void __launch_bounds__(256)
build_k_kernel(const int*   __restrict__ users,
               const int*   __restrict__ pois,
               const float* __restrict__ Dt,
               const float* __restrict__ poi_emb,
               const float* __restrict__ user_emb,
               const float* __restrict__ mean_ptr,
               float*       __restrict__ Kout)
{
    extern __shared__ float smem[];
    float* A_lds = smem;            // 16 x 16   user-embedding tile
    float* S_lds = smem + 256;      // 16 x 4096 S tile (256 KB)

    const int tid  = threadIdx.x;
    const int lane = tid & 31;
    const int wave = tid >> 5;
    const int b0   = blockIdx.x * 16;

    // Gather the 16x16 A tile (user embeddings for this b-tile) into LDS.
    {
        const int row = tid >> 4;          // 0..15
        const int col = tid & 15;          // 0..15
        const int uid = users[b0 + row];
        A_lds[row * 16 + col] = user_emb[uid * DDIM + col];
    }
    __syncthreads();

    // ISA layout, 32-bit 16x4 A / 4x16 B (wave32):
    //   lanes 0-15 : VGPR0=K0, VGPR1=K1 ; lanes 16-31 : VGPR0=K2, VGPR1=K3
    const int m     = lane & 15;
    const int kbase = (lane >> 4) << 1;    // 0 or 2

    v2f a[4];
#pragma unroll
    for (int s = 0; s < 4; ++s) {
        a[s].x = A_lds[m * 16 + 4 * s + kbase];
        a[s].y = A_lds[m * 16 + 4 * s + kbase + 1];
    }

    // 256 column tiles, wave-strided (uniform per wave -> EXEC all ones).
    for (int t = wave; t < NDIM / 16; t += 8) {
        const int nb = t * 16;
        const int n  = nb + m;             // this lane's output column
        v8f c = {};
#pragma unroll
        for (int s = 0; s < 4; ++s) {
            v2f bfrag;                     // B[K][n] = poi_emb[n][K], K pair contiguous
            bfrag.x = poi_emb[n * DDIM + 4 * s + kbase];
            bfrag.y = poi_emb[n * DDIM + 4 * s + kbase + 1];
            c = __builtin_amdgcn_wmma_f32_16x16x4_f32(
                    false, a[s], false, bfrag, (short)0, c, false, false);
        }
        // C layout: VGPR r -> M = r (lanes 0-15) / 8+r (lanes 16-31), N = lane&15
        const int mrow0 = (lane < 16) ? 0 : 8;
#pragma unroll
        for (int r = 0; r < 8; ++r)
            S_lds[(mrow0 + r) * NDIM + n] = c[r];
    }
    __syncthreads();

    // Gather + exp:  K[b,n] = exp(5*S[b, pois[b,n]] - 5*Dn[b,n])
    const float inv_mean = 1.0f / mean_ptr[0];
    for (int e = tid; e < 16 * NDIM; e += 256) {
        const int  bl  = e >> 12;              // local row 0..15
        const int  n   = e & (NDIM - 1);
        const long g   = (long)(b0 + bl) * NDIM + n;
        const int  idx = pois[g];
        const float s  = S_lds[bl * NDIM + idx];
        const float d  = Dt[g];
        Kout[g] = __expf(5.0f * s - 5.0f * d * inv_mean);
    }
}

// ---------------- Phase 2: Sinkhorn -------------------------------------
__global__ void __launch_bounds__(256)
init_v_kernel(float* __restrict__ v)
{
    v[blockIdx.x * 256 + threadIdx.x] = 1.0f;
}

// u[b] = 1 / sum_n K[b,n] * v[n]   (one block per row, tree reduce)
__global__ void __launch_bounds__(256)
sinkhorn_u_kernel(const float* __restrict__ K, const float* __restrict__ v,
                  float* __restrict__ u)
{
    __shared__ float red[256];
    const long rowoff = (long)blockIdx.x * NDIM;
    float sum = 0.0f;
    for (int n = threadIdx.x; n < NDIM; n += 256)
        sum += K[rowoff + n] * v[n];
    red[threadIdx.x] = sum;
    __syncthreads();
    for (int s = 128; s > 0; s >>= 1) {
        if (threadIdx.x < (unsigned)s) red[threadIdx.x] += red[threadIdx.x + s];
        __syncthreads();
    }
    if (threadIdx.x == 0) u[blockIdx.x] = 1.0f / red[0];
}

// pv[ry][j] = sum_{b in 256-row chunk ry} K[b,j]*u[b]  (coalesced along j)
__global__ void __launch_bounds__(256)
sinkhorn_v_partial_kernel(const float* __restrict__ K, const float* __restrict__ u,
                          float* __restrict__ pv)
{
    __shared__ float us[256];
    const int j  = blockIdx.x * 256 + threadIdx.x;
    const int r0 = blockIdx.y * 256;
    us[threadIdx.x] = u[r0 + threadIdx.x];
    __syncthreads();
    float sum = 0.0f;
    for (int i = 0; i < 256; ++i)
        sum += K[(long)(r0 + i) * NDIM + j] * us[i];
    pv[(long)blockIdx.y * NDIM + j] = sum;
}

// v[j] = cap[j] / sum_ry pv[ry][j]   (fixed-order -> deterministic)
__global__ void __launch_bounds__(256)
sinkhorn_v_combine_kernel(const float* __restrict__ pv, const float* __restrict__ cap,
                          float* __restrict__ v)
{
    const int j = blockIdx.x * 256 + threadIdx.x;
    float sum = 0.0f;
#pragma unroll
    for (int r = 0; r < 16; ++r)
        sum += pv[r * NDIM + j];
    v[j] = cap[j] / sum;
}

// ---------------- Phase 3: P = K * u * v (in place in d_out) --------------
__global__ void __launch_bounds__(256)
scale_kernel(float* __restrict__ P, const float* __restrict__ u,
             const float* __restrict__ v)
{
    const long i = (long)blockIdx.x * 256 + threadIdx.x;
    const int  b = (int)(i >> 12);
    const int  n = (int)(i & (NDIM - 1));
    P[i] = P[i] * u[b] * v[n];
}

// ---------------------------------------------------------------------------
extern "C" void kernel_launch(void* const* d_in, const int* in_sizes, int n_in,
                              void* d_out, int out_size, void* d_ws, size_t ws_size,
                              hipStream_t stream)
{
    (void)in_sizes; (void)n_in; (void)out_size; (void)ws_size;

    const int*   users    = (const int*)  d_in[0];   // [B]
    const int*   pois     = (const int*)  d_in[1];   // [B,N]
    const float* Dt       = (const float*)d_in[2];   // [B,N]
    const float* poi_emb  = (const float*)d_in[3];   // [N,16]
    const float* user_emb = (const float*)d_in[4];   // [N_USERS,16]
    const float* cap      = (const float*)d_in[5];   // [N]

    float* K = (float*)d_out;                        // K lives in d_out, scaled in place

    float* wsf      = (float*)d_ws;
    float* partials = wsf;                           // 256
    float* mean     = wsf + 256;                     // 1
    float* u        = wsf + 512;                     // 4096
    float* v        = wsf + 512 + NDIM;              // 4096
    float* pv       = wsf + 512 + 2 * NDIM;          // 16*4096

    // Phase 0: mean(D_tensor)
    mean_partial_kernel<<<256, 256, 0, stream>>>(Dt, partials);
    mean_final_kernel<<<1, 256, 0, stream>>>(partials, mean);

    // Phase 1: build K (WMMA GEMM tile + LDS gather + exp)
    const size_t lds_bytes = (size_t)(256 + 16 * NDIM) * sizeof(float); // ~257 KB < 320 KB
    build_k_kernel<<<BDIM / 16, 256, lds_bytes, stream>>>(
        users, pois, Dt, poi_emb, user_emb, mean, K);

    // Phase 2: Sinkhorn iterations (K stays L2-resident: 64MB << 192MB)
    init_v_kernel<<<NDIM / 256, 256, 0, stream>>>(v);
    for (int it = 0; it < 10; ++it) {
        sinkhorn_u_kernel<<<BDIM, 256, 0, stream>>>(K, v, u);
        sinkhorn_v_partial_kernel<<<dim3(NDIM / 256, BDIM / 256), 256, 0, stream>>>(K, u, pv);
        sinkhorn_v_combine_kernel<<<NDIM / 256, 256, 0, stream>>>(pv, cap, v);
    }

    // Phase 3: P = K * u * v
    scale_kernel<<<(long)BDIM * NDIM / 256, 256, 0, stream>>>(K, u, v);
}